// VQVAE_45277545234647
// MI455X (gfx1250) — compile-verified
//
#include <hip/hip_runtime.h>
#include <math.h>

typedef float v2f __attribute__((ext_vector_type(2)));
typedef float v8f __attribute__((ext_vector_type(8)));

__device__ __forceinline__ v8f wmma4(v2f a, v2f b, v8f c) {
    // V_WMMA_F32_16X16X4_F32 : D = A(16x4,f32) * B(4x16,f32) + C(16x16,f32)
    return __builtin_amdgcn_wmma_f32_16x16x4_f32(
        /*neg_a=*/false, a, /*neg_b=*/false, b,
        /*c_mod=*/(short)0, c, /*reuse_a=*/false, /*reuse_b=*/false);
}

__device__ __forceinline__ float sigm(float x) { return 1.0f / (1.0f + expf(-x)); }

// ============================================================================
// Encoder gated strided conv:  y[b,o,t] = tanh(conv) * sigmoid(gate)
// k=4, s=2, pad=0.  Implicit GEMM: M=O, K=I*4, N=Lout.
// One wave: 32 out-channels (2 M-tiles) x 64 time steps (4 N-tiles), conv+gate.
// K-step of 4 = the 4 taps of one input channel; B fragment is a contiguous
// float2 at x[cin][2t + 2*half].  A/B fragments are double-buffered in
// registers: step k+1's loads issue before step k's 16 WMMAs.
// ============================================================================
__global__ __launch_bounds__(128)
void enc_gated_conv(const float* __restrict__ x,   // [B, I, Lin]
                    const float* __restrict__ cw,  // [O, I, 4]
                    const float* __restrict__ cb,  // [O]
                    const float* __restrict__ gw,  // [O, I, 4]
                    const float* __restrict__ gb,  // [O]
                    float* __restrict__ y,         // [B, O, Lout]
                    int I, int Lin, int Lout, int O)
{
    const int lane   = threadIdx.x & 31;
    const int wave   = threadIdx.x >> 5;
    const int b      = blockIdx.z;
    const int m_base = (blockIdx.y * 4 + wave) * 32;   // 2 M-tiles per wave
    if (m_base >= O) return;
    const int n_base = blockIdx.x * 64;
    const int half = lane >> 4;      // 0: K+{0,1}, 1: K+{2,3}
    const int lrow = lane & 15;

    const int I4 = I * 4;
    const int o0 = m_base + lrow;
    const int o1t = m_base + 16 + lrow;
    const int o1 = (o1t < O) ? o1t : (O - 1);          // clamp; stores guarded

    v8f accC[2][4], accG[2][4];
    const v8f vz = {0.f,0.f,0.f,0.f,0.f,0.f,0.f,0.f};
#pragma unroll
    for (int mt = 0; mt < 2; ++mt)
#pragma unroll
        for (int j = 0; j < 4; ++j) { accC[mt][j] = vz; accG[mt][j] = vz; }

    // clamp tile-overhang time indices so loads stay in-bounds (stores guarded)
    int tj[4];
#pragma unroll
    for (int j = 0; j < 4; ++j) {
        int t = n_base + j * 16 + lrow;
        tj[j] = (t < Lout) ? t : (Lout - 1);
    }

    const float* cw0 = cw + (size_t)o0 * I4 + 2 * half;
    const float* cw1 = cw + (size_t)o1 * I4 + 2 * half;
    const float* gw0 = gw + (size_t)o0 * I4 + 2 * half;
    const float* gw1 = gw + (size_t)o1 * I4 + 2 * half;
    const float* xb  = x  + (size_t)b * I * Lin;

    // ---- prologue: fragments for cin = 0 ----
    v2f aC0 = *(const v2f*)(cw0);
    v2f aC1 = *(const v2f*)(cw1);
    v2f aG0 = *(const v2f*)(gw0);
    v2f aG1 = *(const v2f*)(gw1);
    v2f bc[4];
#pragma unroll
    for (int j = 0; j < 4; ++j) bc[j] = *(const v2f*)(xb + 2 * tj[j] + 2 * half);

#pragma unroll 1
    for (int cin = 0; cin < I; ++cin) {
        const int nx = (cin + 1 < I) ? (cin + 1) : cin;   // last iter: dummy refetch
        // ---- issue next-step loads first (latency hidden under WMMAs) ----
        v2f aC0n = *(const v2f*)(cw0 + nx * 4);
        v2f aC1n = *(const v2f*)(cw1 + nx * 4);
        v2f aG0n = *(const v2f*)(gw0 + nx * 4);
        v2f aG1n = *(const v2f*)(gw1 + nx * 4);
        const float* xn = xb + (size_t)nx * Lin;
        v2f bn[4];
#pragma unroll
        for (int j = 0; j < 4; ++j) bn[j] = *(const v2f*)(xn + 2 * tj[j] + 2 * half);

        // ---- 16 WMMAs on current fragments ----
#pragma unroll
        for (int j = 0; j < 4; ++j) {
            accC[0][j] = wmma4(aC0, bc[j], accC[0][j]);
            accG[0][j] = wmma4(aG0, bc[j], accG[0][j]);
            accC[1][j] = wmma4(aC1, bc[j], accC[1][j]);
            accG[1][j] = wmma4(aG1, bc[j], accG[1][j]);
        }
        aC0 = aC0n; aC1 = aC1n; aG0 = aG0n; aG1 = aG1n;
#pragma unroll
        for (int j = 0; j < 4; ++j) bc[j] = bn[j];
    }

#pragma unroll
    for (int mt = 0; mt < 2; ++mt) {
#pragma unroll
        for (int j = 0; j < 4; ++j) {
            int t = n_base + j * 16 + lrow;
            if (t >= Lout) continue;
#pragma unroll
            for (int r = 0; r < 8; ++r) {
                int ch = m_base + mt * 16 + r + 8 * half;
                if (ch >= O) continue;
                float c = accC[mt][j][r] + cb[ch];
                float g = accG[mt][j][r] + gb[ch];
                y[((size_t)b * O + ch) * Lout + t] = tanhf(c) * sigm(g);
            }
        }
    }
}

// ============================================================================
// Latent 1x1 conv: z[b,d,t] = sum_c lat_w[d,c] * x[b,c,t] + lat_b[d]
// GEMM: M=64, K=256, N=Lz.  K-step of 4 channels, register double buffer.
// ============================================================================
__global__ __launch_bounds__(128)
void latent_conv(const float* __restrict__ x,    // [B, 256, Lz]
                 const float* __restrict__ w,    // [64, 256]
                 const float* __restrict__ bias, // [64]
                 float* __restrict__ z,          // [B, 64, Lz]
                 int Lz)
{
    const int lane = threadIdx.x & 31;
    const int wave = threadIdx.x >> 5;
    const int b = blockIdx.z;
    const int m_base = wave * 16;             // 4 waves cover 64 channels
    const int n_base = blockIdx.x * 64;
    const int half = lane >> 4, lrow = lane & 15;

    v8f acc[4];
    const v8f vz = {0.f,0.f,0.f,0.f,0.f,0.f,0.f,0.f};
#pragma unroll
    for (int j = 0; j < 4; ++j) acc[j] = vz;

    int tj[4];
#pragma unroll
    for (int j = 0; j < 4; ++j) {
        int t = n_base + j * 16 + lrow;
        tj[j] = (t < Lz) ? t : (Lz - 1);
    }

    const float* wp = w + (size_t)(m_base + lrow) * 256 + 2 * half;
    const float* xb = x + (size_t)b * 256 * Lz;

    v2f ac = *(const v2f*)(wp);
    v2f bc[4];
    {
        const float* x0 = xb + (size_t)(2 * half) * Lz;
#pragma unroll
        for (int j = 0; j < 4; ++j) bc[j] = (v2f){ x0[tj[j]], x0[Lz + tj[j]] };
    }

#pragma unroll 1
    for (int c0 = 0; c0 < 256; c0 += 4) {
        const int nx = (c0 + 4 < 256) ? (c0 + 4) : c0;
        v2f an = *(const v2f*)(wp + nx);
        const float* xn = xb + (size_t)(nx + 2 * half) * Lz;
        v2f bn[4];
#pragma unroll
        for (int j = 0; j < 4; ++j) bn[j] = (v2f){ xn[tj[j]], xn[Lz + tj[j]] };

#pragma unroll
        for (int j = 0; j < 4; ++j) acc[j] = wmma4(ac, bc[j], acc[j]);

        ac = an;
#pragma unroll
        for (int j = 0; j < 4; ++j) bc[j] = bn[j];
    }

#pragma unroll
    for (int j = 0; j < 4; ++j) {
        int t = n_base + j * 16 + lrow;
        if (t >= Lz) continue;
#pragma unroll
        for (int r = 0; r < 8; ++r) {
            int ch = m_base + r + 8 * half;
            z[((size_t)b * 64 + ch) * Lz + t] = acc[j][r] + bias[ch];
        }
    }
}

// ============================================================================
// Vector quantization: per (b,t) argmin_k (||c_k||^2 - 2 z.c_k)   (||z||^2 const)
// Codebook streamed through LDS in 64-code (16 KB) chunks.
// ============================================================================
__global__ __launch_bounds__(256)
void vq_kernel(const float* __restrict__ z,        // [B, 64, Tz]
               const float* __restrict__ codebook, // [512, 64]
               float* __restrict__ zemb,           // [B, 64, Tz]
               float* __restrict__ qbuf,           // [B, 64, Tz]
               int Tz)
{
    __shared__ float cbs[64 * 64];
    const int tid = threadIdx.x;
    const int gid = blockIdx.x * 256 + tid;
    const bool valid = gid < 8 * Tz;
    const int b = gid / Tz;
    const int t = gid - b * Tz;

    float zv[64];
    if (valid) {
#pragma unroll 8
        for (int d = 0; d < 64; ++d) zv[d] = z[((size_t)b * 64 + d) * Tz + t];
    }

    float best = 3.4e38f;
    int bidx = 0;
    for (int chunk = 0; chunk < 8; ++chunk) {
        __syncthreads();
        for (int i = tid; i < 64 * 64; i += 256) cbs[i] = codebook[chunk * 64 * 64 + i];
        __syncthreads();
        if (valid) {
            for (int c = 0; c < 64; ++c) {
                const float* cp = &cbs[c * 64];
                float dot = 0.f, cc = 0.f;
#pragma unroll 8
                for (int d = 0; d < 64; ++d) { float cv = cp[d]; dot += zv[d] * cv; cc += cv * cv; }
                float score = cc - 2.f * dot;
                if (score < best) { best = score; bidx = chunk * 64 + c; }
            }
        }
    }

    if (valid) {
        const float* q = codebook + (size_t)bidx * 64;
#pragma unroll 8
        for (int d = 0; d < 64; ++d) {
            float qv = q[d];
            zemb[((size_t)b * 64 + d) * Tz + t] = qv;
            qbuf[((size_t)b * 64 + d) * Tz + t] = qv;
        }
    }
}

// ============================================================================
// Speaker conditioning: cond[b,o] = spk_emb[spk[b]] . lw[o,:] + lb[o]
// ============================================================================
__global__ void cond_kernel(const int* __restrict__ spk,
                            const float* __restrict__ spk_emb, // [109,64]
                            const float* __restrict__ lw,      // [O,64]
                            const float* __restrict__ lb,      // [O]
                            float* __restrict__ cond,          // [B,O]
                            int O)
{
    int o = blockIdx.x * blockDim.x + threadIdx.x;
    int b = blockIdx.y;
    if (o >= O) return;
    const float* h = spk_emb + (size_t)spk[b] * 64;
    const float* w = lw + (size_t)o * 64;
    float acc = lb[o];
#pragma unroll 8
    for (int d = 0; d < 64; ++d) acc += h[d] * w[d];
    cond[b * O + o] = acc;
}

// ============================================================================
// Decoder gated transpose conv (k=4, s=2, op=0) + speaker cond.
// Split by output parity p: t = 2u + p, u in [0, Lin], Lout = 2*Lin + 2.
//   out[o, 2u+p] = sum_i sum_{j in {0,1}} w[i][o][p+2j] * x[i][u-j]
// GEMM per parity: M=O, K=2*I (q = 2i+j), N = u.  Per lane: i = i0 + half, j = r.
// 2 M-tiles per wave; boundary handled with hoisted clamped indices + 0/1
// multiplicative masks (no per-iteration selects); register double buffer.
// ============================================================================
__global__ __launch_bounds__(128)
void dec_gated_tconv(const float* __restrict__ x,   // [B, I, Lin]
                     const float* __restrict__ cw,  // [I, O, 4] (torch layout)
                     const float* __restrict__ cb,  // [O]
                     const float* __restrict__ gw,  // [I, O, 4]
                     const float* __restrict__ gb,  // [O]
                     const float* __restrict__ cond,// [B, O]
                     float* __restrict__ y,         // [B, O, 2*Lin+2]
                     int I, int Lin, int O)
{
    const int lane = threadIdx.x & 31;
    const int wave = threadIdx.x >> 5;
    const int bz = blockIdx.z;
    const int b = bz >> 1;
    const int p = bz & 1;
    const int m_base = (blockIdx.y * 4 + wave) * 32;   // 2 M-tiles per wave
    if (m_base >= O) return;
    const int u_base = blockIdx.x * 64;
    const int Lout = 2 * Lin + 2;
    const int half = lane >> 4, lrow = lane & 15;
    const int o0 = m_base + lrow;
    const int o1t = m_base + 16 + lrow;
    const int o1 = (o1t < O) ? o1t : (O - 1);

    v8f accC[2][4], accG[2][4];
    const v8f vz = {0.f,0.f,0.f,0.f,0.f,0.f,0.f,0.f};
#pragma unroll
    for (int mt = 0; mt < 2; ++mt)
#pragma unroll
        for (int j = 0; j < 4; ++j) { accC[mt][j] = vz; accG[mt][j] = vz; }

    // hoisted per-N-tile clamped source indices + masks (u fixed across K loop)
    int u0c[4], u1c[4];
    float msk0[4], msk1[4];
#pragma unroll
    for (int j = 0; j < 4; ++j) {
        int u = u_base + j * 16 + lrow;
        u0c[j]  = (u < Lin) ? u : (Lin - 1);                    // src = u
        msk0[j] = (u < Lin) ? 1.0f : 0.0f;
        int um1 = u - 1;                                        // src = u-1
        u1c[j]  = (um1 < 0) ? 0 : ((um1 < Lin) ? um1 : (Lin - 1));
        msk1[j] = (u >= 1 && u <= Lin) ? 1.0f : 0.0f;
    }

    const float* xb = x + (size_t)b * I * Lin;
    const size_t O4 = (size_t)O * 4;

    // ---- prologue: fragments for i0 = 0 (i = half) ----
    int i = half;
    const float* wc0 = cw + (size_t)i * O4 + (size_t)o0 * 4 + p;
    const float* wc1 = cw + (size_t)i * O4 + (size_t)o1 * 4 + p;
    const float* wg0 = gw + (size_t)i * O4 + (size_t)o0 * 4 + p;
    const float* wg1 = gw + (size_t)i * O4 + (size_t)o1 * 4 + p;
    v2f aC0 = { wc0[0], wc0[2] };     // j=0 -> tap p, j=1 -> tap p+2
    v2f aC1 = { wc1[0], wc1[2] };
    v2f aG0 = { wg0[0], wg0[2] };
    v2f aG1 = { wg1[0], wg1[2] };
    v2f bc[4];
    {
        const float* xc = xb + (size_t)i * Lin;
#pragma unroll
        for (int j = 0; j < 4; ++j)
            bc[j] = (v2f){ xc[u0c[j]] * msk0[j], xc[u1c[j]] * msk1[j] };
    }

#pragma unroll 1
    for (int i0 = 0; i0 < I; i0 += 2) {
        const int nxi = ((i0 + 2 < I) ? (i0 + 2) : i0) + half; // last iter: dummy
        // ---- issue next-step loads ----
        const float* nwc0 = cw + (size_t)nxi * O4 + (size_t)o0 * 4 + p;
        const float* nwc1 = cw + (size_t)nxi * O4 + (size_t)o1 * 4 + p;
        const float* nwg0 = gw + (size_t)nxi * O4 + (size_t)o0 * 4 + p;
        const float* nwg1 = gw + (size_t)nxi * O4 + (size_t)o1 * 4 + p;
        v2f aC0n = { nwc0[0], nwc0[2] };
        v2f aC1n = { nwc1[0], nwc1[2] };
        v2f aG0n = { nwg0[0], nwg0[2] };
        v2f aG1n = { nwg1[0], nwg1[2] };
        const float* xn = xb + (size_t)nxi * Lin;
        v2f bn[4];
#pragma unroll
        for (int j = 0; j < 4; ++j)
            bn[j] = (v2f){ xn[u0c[j]] * msk0[j], xn[u1c[j]] * msk1[j] };

        // ---- 16 WMMAs on current fragments ----
#pragma unroll
        for (int j = 0; j < 4; ++j) {
            accC[0][j] = wmma4(aC0, bc[j], accC[0][j]);
            accG[0][j] = wmma4(aG0, bc[j], accG[0][j]);
            accC[1][j] = wmma4(aC1, bc[j], accC[1][j]);
            accG[1][j] = wmma4(aG1, bc[j], accG[1][j]);
        }
        aC0 = aC0n; aC1 = aC1n; aG0 = aG0n; aG1 = aG1n;
#pragma unroll
        for (int j = 0; j < 4; ++j) bc[j] = bn[j];
    }

#pragma unroll
    for (int mt = 0; mt < 2; ++mt) {
#pragma unroll
        for (int j = 0; j < 4; ++j) {
            int u = u_base + j * 16 + lrow;
            if (u > Lin) continue;
            int t = 2 * u + p;
#pragma unroll
            for (int r = 0; r < 8; ++r) {
                int ch = m_base + mt * 16 + r + 8 * half;
                if (ch >= O) continue;
                float cd = cond[b * O + ch];
                float c = accC[mt][j][r] + cb[ch] + cd;
                float g = accG[mt][j][r] + gb[ch] + cd;
                y[((size_t)b * O + ch) * Lout + t] = tanhf(c) * sigm(g);
            }
        }
    }
}

// ============================================================================
extern "C" void kernel_launch(void* const* d_in, const int* in_sizes, int n_in,
                              void* d_out, int out_size, void* d_ws, size_t ws_size,
                              hipStream_t stream)
{
    (void)in_sizes; (void)n_in; (void)out_size; (void)ws_size;

    const float* x    = (const float*)d_in[0];
    const int*   spk  = (const int*)  d_in[1];
    const float* e1cw = (const float*)d_in[2];
    const float* e1cb = (const float*)d_in[3];
    const float* e1gw = (const float*)d_in[4];
    const float* e1gb = (const float*)d_in[5];
    const float* e2cw = (const float*)d_in[6];
    const float* e2cb = (const float*)d_in[7];
    const float* e2gw = (const float*)d_in[8];
    const float* e2gb = (const float*)d_in[9];
    const float* e3cw = (const float*)d_in[10];
    const float* e3cb = (const float*)d_in[11];
    const float* e3gw = (const float*)d_in[12];
    const float* e3gb = (const float*)d_in[13];
    const float* latw = (const float*)d_in[14];
    const float* latb = (const float*)d_in[15];
    const float* cbk  = (const float*)d_in[16];
    const float* semb = (const float*)d_in[17];
    const float* g1cw = (const float*)d_in[18];
    const float* g1cb = (const float*)d_in[19];
    const float* g1gw = (const float*)d_in[20];
    const float* g1gb = (const float*)d_in[21];
    const float* g1lw = (const float*)d_in[22];
    const float* g1lb = (const float*)d_in[23];
    const float* g2cw = (const float*)d_in[24];
    const float* g2cb = (const float*)d_in[25];
    const float* g2gw = (const float*)d_in[26];
    const float* g2gb = (const float*)d_in[27];
    const float* g2lw = (const float*)d_in[28];
    const float* g2lb = (const float*)d_in[29];
    const float* g3cw = (const float*)d_in[30];
    const float* g3cb = (const float*)d_in[31];
    const float* g3gw = (const float*)d_in[32];
    const float* g3gb = (const float*)d_in[33];
    const float* g3lw = (const float*)d_in[34];
    const float* g3lb = (const float*)d_in[35];

    // lengths: 8192 -> 4095 -> 2046 -> 1022 (enc); 1022 -> 2046 -> 4094 -> 8190 (dec)
    const int L0 = 8192, L1 = 4095, L2 = 2046, L3 = 1022;
    const int D2in = 2046, D3in = 4094;

    float* ws = (float*)d_ws;
    // workspace layout (floats), dec buffers alias freed enc buffers
    float* enc1  = ws;                       // 8*256*4095 = 8,386,560
    float* enc2  = ws + 8386560;             // 8*256*2046 = 4,190,208
    float* enc3  = ws + 12576768;            // 8*256*1022 = 2,093,056
    float* qbuf  = ws + 14669824;            // 8*64*1022  =   523,264
    float* cond1 = ws + 15193088;            // 8*256
    float* cond2 = cond1 + 2048;             // 8*256
    float* cond3 = cond2 + 2048;             // 8*80
    float* dec1  = enc2;                     // 8*256*2046 (enc2 dead by then)
    float* dec2  = enc1;                     // 8*256*4094 <= enc1 slot

    float* out       = (float*)d_out;
    float* out_main  = out;                  // [8, 80, 8190] = 5,241,600
    float* out_zemb  = out + 5241600;        // [8, 64, 1022] =   523,264
    float* out_z     = out + 5241600 + 523264;

    const dim3 B128(128);

    // ---- encoder (block = 4 waves x 32 ch = 128 ch -> grid.y = O/128) ----
    enc_gated_conv<<<dim3(64, 2, 8), B128, 0, stream>>>(x,    e1cw, e1cb, e1gw, e1gb, enc1,  80, L0, L1, 256);
    enc_gated_conv<<<dim3(32, 2, 8), B128, 0, stream>>>(enc1, e2cw, e2cb, e2gw, e2gb, enc2, 256, L1, L2, 256);
    enc_gated_conv<<<dim3(16, 2, 8), B128, 0, stream>>>(enc2, e3cw, e3cb, e3gw, e3gb, enc3, 256, L2, L3, 256);

    // ---- latent + VQ ----
    latent_conv<<<dim3(16, 1, 8), B128, 0, stream>>>(enc3, latw, latb, out_z, L3);
    vq_kernel<<<dim3((8 * L3 + 255) / 256), dim3(256), 0, stream>>>(out_z, cbk, out_zemb, qbuf, L3);

    // ---- speaker conditioning ----
    cond_kernel<<<dim3(1, 8), dim3(256), 0, stream>>>(spk, semb, g1lw, g1lb, cond1, 256);
    cond_kernel<<<dim3(1, 8), dim3(256), 0, stream>>>(spk, semb, g2lw, g2lb, cond2, 256);
    cond_kernel<<<dim3(1, 8), dim3(256), 0, stream>>>(spk, semb, g3lw, g3lb, cond3,  80);

    // ---- decoder (grid.z = B * 2 parities); u-tiles = ceil((Lin+1)/64) ----
    dec_gated_tconv<<<dim3(16, 2, 16), B128, 0, stream>>>(qbuf, g1cw, g1cb, g1gw, g1gb, cond1, dec1,      64, L3,   256);
    dec_gated_tconv<<<dim3(32, 2, 16), B128, 0, stream>>>(dec1, g2cw, g2cb, g2gw, g2gb, cond2, dec2,     256, D2in, 256);
    dec_gated_tconv<<<dim3(64, 1, 16), B128, 0, stream>>>(dec2, g3cw, g3cb, g3gw, g3gb, cond3, out_main, 256, D3in,  80);
}